// RosaSeqLayer_77094662963959
// MI455X (gfx1250) — compile-verified
//
#include <hip/hip_runtime.h>
#include <stdint.h>

#define B_    4
#define T_    4096
#define D_    1024
#define H_    32
#define NX_   12
#define NV_   32
#define KD    1024
#define NWX   (H_ * NX_)   /* 384  */
#define NWV   (H_ * NV_)   /* 1024 */
#define M_TOT (B_ * T_)    /* 16384 */

#define SCAN_STREAMS 8     /* streams (tables) per scan block: 8 * 16KB = 128KB LDS */

typedef float v2f  __attribute__((ext_vector_type(2)));
typedef float v8f  __attribute__((ext_vector_type(8)));
typedef unsigned int u32x4 __attribute__((ext_vector_type(4)));
typedef int   i32x4 __attribute__((ext_vector_type(4)));
typedef int   i32x8 __attribute__((ext_vector_type(8)));

#if __has_builtin(__builtin_amdgcn_tensor_load_to_lds)
#define ROSA_TDM 1
#else
#define ROSA_TDM 0
#endif

#if ROSA_TDM
/* TDM pad: 1 DWORD inserted every 256 DWORDs -> row stride 1028 DWORDs.
 * (r,k) lives at DWORD offset r*1028 + k + (k>>8).  Bank for lane l reading
 * column kk: (l*1028 + c) % 64 = (4l + c) % 64 -> 16 distinct banks; upper
 * half-wave reads kk+2 -> disjoint odd-offset bank set.                      */
#define A_ROW_STRIDE 1028
#define A_KOFF(kk)   ((kk) + ((kk) >> 8))
#else
#define A_ROW_STRIDE 1032
#define A_KOFF(kk)   (kk)
#endif
#define LDS_A_BYTES  ((16 * 1032 + 32) * 4)   /* covers both layouts */

__device__ __forceinline__ unsigned ballot32(bool p) {
#if __has_builtin(__builtin_amdgcn_ballot_w32)
  return __builtin_amdgcn_ballot_w32(p);
#else
  return (unsigned)__ballot(p);
#endif
}

// D = A(16x4 f32) * B(4x16 f32) + C, f32 accum. 8-arg CDNA5 form.
__device__ __forceinline__ v8f wmma_f32_k4(v2f a, v2f b, v8f c) {
  return __builtin_amdgcn_wmma_f32_16x16x4_f32(false, a, false, b, (short)0, c,
                                               false, false);
}

// ---------------------------------------------------------------------------
// Stage the 16 x 1024 f32 A-tile of x into LDS.
// TDM path: wave 0 issues one tensor_load_to_lds (2D descriptor, pad enabled),
// waits TENSORcnt, then the workgroup barrier publishes the tile.
// ---------------------------------------------------------------------------
__device__ __forceinline__ void stage_a_tile(const float* __restrict__ gsrc,
                                             float* xs, int tid, int wave) {
#if ROSA_TDM
  if (wave == 0) {
    const unsigned long long ga = (unsigned long long)(uintptr_t)gsrc;
    const unsigned lds_off = (unsigned)(uintptr_t)xs;   // low 32 bits = LDS byte addr
    u32x4 g0;
    g0.x = 1u;                                          // count=1, user mode
    g0.y = lds_off;                                     // lds_addr [63:32]
    g0.z = (unsigned)(ga & 0xFFFFFFFFu);                // global_addr [95:64]
    g0.w = (unsigned)((ga >> 32) & 0x01FFFFFFu)         // global_addr [120:96]
         | (2u << 30);                                  // type=2 ("image")
    i32x8 g1;
    g1[0] = (2 << 16)        // data_size = 4 bytes
          | (1 << 20)        // pad_enable
          | (7 << 22)        // pad_interval code 7 = 256 DWORDs
          | (0 << 25);       // pad_amount  code 0 = 1 DWORD
    g1[1] = (int)(1024u << 16);   // tensor_dim0[15:0] in bits [63:48]
    g1[2] = (int)(16u << 16);     // tensor_dim1[15:0] in bits [95:80]
    g1[3] = (int)(1024u << 16);   // tile_dim0 in bits [127:112]
    g1[4] = 16;                   // tile_dim1 in bits [143:128]
    g1[5] = 1024;                 // tensor_dim0_stride[31:0]
    g1[6] = 0;
    g1[7] = 0;
    const i32x4 z4 = {0, 0, 0, 0};
#if __clang_major__ >= 23
    const i32x8 z8 = {0, 0, 0, 0, 0, 0, 0, 0};
    __builtin_amdgcn_tensor_load_to_lds(g0, g1, z4, z4, z8, 0);
#else
    __builtin_amdgcn_tensor_load_to_lds(g0, g1, z4, z4, 0);
#endif
    __builtin_amdgcn_s_wait_tensorcnt(0);
  }
  (void)tid;
#else
  for (int i = tid; i < 16 * (KD / 4); i += 256) {
    const int r = i >> 8;
    const int c = (i & 255) << 2;
    const float4 v = *(const float4*)(gsrc + (size_t)r * KD + c);
    *(float4*)(xs + r * A_ROW_STRIDE + c) = v;
  }
  (void)wave;
#endif
  __syncthreads();
}

// ---------------------------------------------------------------------------
// GEMM + sign-pack for 12-bit keys.  Block = 16-row M-tile, wave = 48-col strip
// (heads 4w..4w+3).  A tile staged by TDM into LDS, B streamed from L2.
// ---------------------------------------------------------------------------
__global__ void __launch_bounds__(256)
rosa_gemm_kx(const float* __restrict__ x, const float* __restrict__ wx,
             unsigned* __restrict__ kx) {
  extern __shared__ float xs[];
  const int tid  = threadIdx.x;
  const int wave = tid >> 5;
  const int lane = tid & 31;
  const int l    = lane & 15;
  const int half = lane >> 4;
  const int m0   = blockIdx.x * 16;

  stage_a_tile(x + (size_t)m0 * KD, xs, tid, wave);

  const int n0 = wave * 48;                    // strip of 3 N-tiles = 4 heads
  v8f acc0 = {}, acc1 = {}, acc2 = {};
  const float* arow = xs + l * A_ROW_STRIDE;

  for (int k0 = 0; k0 < KD; k0 += 4) {
    const int kk = k0 + 2 * half;              // lanes 0-15: K=k0..k0+1, 16-31: k0+2..k0+3
    const int ko = A_KOFF(kk);
    v2f a;  a.x = arow[ko];  a.y = arow[ko + 1];
    const float* wp0 = wx + (size_t)kk       * NWX + n0 + l;
    const float* wp1 = wx + (size_t)(kk + 1) * NWX + n0 + l;
    v2f b0; b0.x = wp0[0];  b0.y = wp1[0];
    v2f b1; b1.x = wp0[16]; b1.y = wp1[16];
    v2f b2; b2.x = wp0[32]; b2.y = wp1[32];
    acc0 = wmma_f32_k4(a, b0, acc0);
    acc1 = wmma_f32_k4(a, b1, acc1);
    acc2 = wmma_f32_k4(a, b2, acc2);
  }

  // C layout: VGPR r -> rows (m0+r, m0+8+r); ballot low16 = row r, high16 = row r+8.
#pragma unroll
  for (int r = 0; r < 8; ++r) {
    const unsigned s0 = ballot32(acc0[r] > 0.0f);
    const unsigned s1 = ballot32(acc1[r] > 0.0f);
    const unsigned s2 = ballot32(acc2[r] > 0.0f);
    if (lane == 0) {
#pragma unroll
      for (int hr = 0; hr < 2; ++hr) {
        const unsigned u0 = (hr ? (s0 >> 16) : s0) & 0xFFFFu;
        const unsigned u1 = (hr ? (s1 >> 16) : s1) & 0xFFFFu;
        const unsigned u2 = (hr ? (s2 >> 16) : s2) & 0xFFFFu;
        const int row = m0 + r + 8 * hr;       // = b*T + t
        const int bb  = row >> 12;
        const int tt  = row & (T_ - 1);
        const unsigned k0v =  u0 & 0xFFFu;
        const unsigned k1v = (u0 >> 12) | ((u1 & 0xFFu) << 4);
        const unsigned k2v = (u1 >> 8)  | ((u2 & 0xFu) << 8);
        const unsigned k3v =  u2 >> 4;
        const int hb = wave * 4;
        kx[((size_t)(bb * H_ + hb + 0)) * T_ + tt] = k0v;
        kx[((size_t)(bb * H_ + hb + 1)) * T_ + tt] = k1v;
        kx[((size_t)(bb * H_ + hb + 2)) * T_ + tt] = k2v;
        kx[((size_t)(bb * H_ + hb + 3)) * T_ + tt] = k3v;
      }
    }
  }
}

// ---------------------------------------------------------------------------
// GEMM + sign-pack for 32-bit values.  Wave = 128-col strip = heads 4w..4w+3.
// ---------------------------------------------------------------------------
__global__ void __launch_bounds__(256)
rosa_gemm_kv(const float* __restrict__ x, const float* __restrict__ wv,
             unsigned* __restrict__ kv) {
  extern __shared__ float xs[];
  const int tid  = threadIdx.x;
  const int wave = tid >> 5;
  const int lane = tid & 31;
  const int l    = lane & 15;
  const int half = lane >> 4;
  const int m0   = blockIdx.x * 16;

  stage_a_tile(x + (size_t)m0 * KD, xs, tid, wave);

  const int n0 = wave * 128;                   // 8 N-tiles = 4 heads of 32 cols
  v8f acc[8];
#pragma unroll
  for (int i = 0; i < 8; ++i) acc[i] = (v8f){};
  const float* arow = xs + l * A_ROW_STRIDE;

  for (int k0 = 0; k0 < KD; k0 += 4) {
    const int kk = k0 + 2 * half;
    const int ko = A_KOFF(kk);
    v2f a;  a.x = arow[ko];  a.y = arow[ko + 1];
    const float* wp0 = wv + (size_t)kk       * NWV + n0 + l;
    const float* wp1 = wv + (size_t)(kk + 1) * NWV + n0 + l;
#pragma unroll
    for (int tn = 0; tn < 8; ++tn) {
      v2f b; b.x = wp0[tn * 16]; b.y = wp1[tn * 16];
      acc[tn] = wmma_f32_k4(a, b, acc[tn]);
    }
  }

#pragma unroll
  for (int p = 0; p < 4; ++p) {
    const int h = wave * 4 + p;
#pragma unroll
    for (int r = 0; r < 8; ++r) {
      const unsigned b0 = ballot32(acc[2 * p + 0][r] > 0.0f);
      const unsigned b1 = ballot32(acc[2 * p + 1][r] > 0.0f);
      if (lane == 0) {
        const int row_lo = m0 + r, row_hi = m0 + 8 + r;
        const unsigned vlo = (b0 & 0xFFFFu) | (b1 << 16);
        const unsigned vhi = (b0 >> 16) | (b1 & 0xFFFF0000u);
        kv[((size_t)((row_lo >> 12) * H_ + h)) * T_ + (row_lo & (T_ - 1))] = vlo;
        kv[((size_t)((row_hi >> 12) * H_ + h)) * T_ + (row_hi & (T_ - 1))] = vhi;
      }
    }
  }
}

// ---------------------------------------------------------------------------
// Sequential associative-memory scan.  128 streams -> 16 blocks x 8 lanes,
// each lane owns a 4096-entry table in LDS.  8-step batches: the next load
// address never depends on the previous load's data, only LDS in-order
// semantics, so ds ops pipeline without per-step round trips.
// ---------------------------------------------------------------------------
__global__ void __launch_bounds__(32)
rosa_scan(const unsigned* __restrict__ kx, const unsigned* __restrict__ kv,
          unsigned* __restrict__ y) {
  extern __shared__ unsigned table[];          // SCAN_STREAMS * 4096
  const int lane = threadIdx.x;
  for (int i = lane; i < SCAN_STREAMS * 4096; i += 32) table[i] = 0u;
  __syncthreads();
  if (lane >= SCAN_STREAMS) return;

  const int stream = blockIdx.x * SCAN_STREAMS + lane;   // 0..127 = b*H + h
  const size_t base = (size_t)stream * T_;
  unsigned* tb = table + lane * 4096;

  for (int t = 0; t < T_; t += 8) {
    unsigned kb[8], vb[8], yb[8];
#pragma unroll
    for (int i = 0; i < 8; ++i) { kb[i] = kx[base + t + i]; vb[i] = kv[base + t + i]; }
#pragma unroll
    for (int i = 0; i < 8; ++i) { yb[i] = tb[kb[i]]; tb[kb[i]] = vb[i]; }
#pragma unroll
    for (int i = 0; i < 8; ++i) y[base + t + i] = yb[i];
  }
}

// ---------------------------------------------------------------------------
// Expand recalled 32-bit values into the [B,T,D] f32 output (write-bound).
// ---------------------------------------------------------------------------
__global__ void __launch_bounds__(256)
rosa_expand(const unsigned* __restrict__ y, const float* __restrict__ emb0,
            const float* __restrict__ emb1, float* __restrict__ out) {
  const size_t idx = (size_t)blockIdx.x * 256 + threadIdx.x;  // over B*T*D/4
  const int    dq  = (int)(idx & (D_ / 4 - 1));               // 0..255
  const size_t row = idx >> 8;                                // b*T + t
  const int d  = dq << 2;
  const int h  = d >> 5;
  const int j  = d & 31;
  const int bb = (int)(row >> 12);
  const int tt = (int)(row & (T_ - 1));
  const unsigned yv = y[((size_t)(bb * H_ + h)) * T_ + tt];
  float4 o;
  o.x = ((yv >> (j + 0)) & 1u) ? emb1[d + 0] : emb0[d + 0];
  o.y = ((yv >> (j + 1)) & 1u) ? emb1[d + 1] : emb0[d + 1];
  o.z = ((yv >> (j + 2)) & 1u) ? emb1[d + 2] : emb0[d + 2];
  o.w = ((yv >> (j + 3)) & 1u) ? emb1[d + 3] : emb0[d + 3];
  *(float4*)(out + row * D_ + d) = o;
}

// ---------------------------------------------------------------------------
extern "C" void kernel_launch(void* const* d_in, const int* in_sizes, int n_in,
                              void* d_out, int out_size, void* d_ws, size_t ws_size,
                              hipStream_t stream) {
  const float* x    = (const float*)d_in[0];
  const float* wx   = (const float*)d_in[1];
  const float* wv   = (const float*)d_in[2];
  const float* emb0 = (const float*)d_in[3];
  const float* emb1 = (const float*)d_in[4];
  (void)in_sizes; (void)n_in; (void)out_size; (void)ws_size;

  unsigned* kx = (unsigned*)d_ws;                       // 2 MB
  unsigned* kv = kx + (size_t)B_ * H_ * T_;             // 2 MB
  unsigned* yb = kv + (size_t)B_ * H_ * T_;             // 2 MB
  float*    out = (float*)d_out;

  const size_t lds_a = (size_t)LDS_A_BYTES;                            // ~66 KB
  const size_t lds_s = (size_t)SCAN_STREAMS * 4096 * sizeof(unsigned); // 128 KB

  rosa_gemm_kx<<<dim3(M_TOT / 16), 256, lds_a, stream>>>(x, wx, kx);
  rosa_gemm_kv<<<dim3(M_TOT / 16), 256, lds_a, stream>>>(x, wv, kv);
  rosa_scan  <<<dim3((B_ * H_) / SCAN_STREAMS), 32, lds_s, stream>>>(kx, kv, yb);
  rosa_expand<<<dim3((B_ * T_ * D_ / 4) / 256), 256, 0, stream>>>(yb, emb0, emb1, out);
}